// CRF_17025250361455
// MI455X (gfx1250) — compile-verified
//
#include <hip/hip_runtime.h>
#include <hip/hip_bf16.h>

// CRF forward (scaled forward algorithm) as f32 WMMA GEMM + TDM streaming:
//   score_t = h_t + c + log( exp(score - c) @ W ),  W = exp(trans^T)
// - Matrix work on V_WMMA_F32_16X16X4_F32 (full f32: 512-step recurrence stays tight).
// - h tiles (64 rows x 48 f32 per block per step) streamed into LDS one step
//   ahead by the Tensor Data Mover (tensor_load_to_lds, TENSORcnt), double
//   buffered; TDM pad feature (1 dword per 16) gives odd LDS row stride 51 ->
//   conflict-free blend reads.

typedef __attribute__((ext_vector_type(2))) float v2f;
typedef __attribute__((ext_vector_type(8))) float v8f;
typedef __attribute__((ext_vector_type(4))) unsigned int v4ui;
typedef __attribute__((ext_vector_type(8))) int v8si;
typedef __attribute__((ext_vector_type(4))) int v4si;

#define B_  1024
#define T_  512
#define L_  48
#define NEG_ (-10000.0f)
#define BOS_ 1

#define ROWS_PER_WAVE  16
#define WAVES_PER_BLK  4
#define ROWS_PER_BLK   (ROWS_PER_WAVE * WAVES_PER_BLK)   // 64
#define SC_STRIDE      49            // odd stride: conflict-free score banks
#define HROW_STRIDE    51            // 48 + 3 TDM pad dwords (pad 1 per 16)
#define HBUF_DW        (ROWS_PER_BLK * HROW_STRIDE)      // 3264 dwords / buffer

// Issue one TDM 2D tile load: 64 rows x 48 f32 of h[:, t, :] -> LDS hbuf.
// Must be called by a full wave (EXEC ignored by TDM); args are uniform.
__device__ __forceinline__ void tdm_load_h_tile(const float* h, int block_row0,
                                                int t, float* lds_dst) {
  const unsigned long long gaddr =
      (unsigned long long)(const void*)h +
      ((unsigned long long)block_row0 * T_ + (unsigned long long)t) * (L_ * 4ull);
  const unsigned int lds_off = (unsigned int)(unsigned long long)lds_dst;

  v4ui g0;
  g0.x = 1u;                                   // count=1 (valid), user mode
  g0.y = lds_off;                              // lds_addr (bytes)
  g0.z = (unsigned int)gaddr;                  // global_addr[31:0]
  g0.w = (unsigned int)((gaddr >> 32) & 0x01FFFFFFull) | (2u << 30); // [56:32]+type=2

  v8si g1;
  g1.s0 = (2 << 16)      // data_size = 4 bytes
        | (1 << 20)      // pad_enable
        | (3 << 22)      // pad_interval: 16 dwords
        | (0 << 25);     // pad_amount: 1 dword  -> LDS row stride 51 (odd)
  g1.s1 = (L_ << 16);            // tensor_dim0 = 48 (bits 63:48)
  g1.s2 = (ROWS_PER_BLK << 16);  // tensor_dim0 hi=0; tensor_dim1 = 64 (bits 95:80)
  g1.s3 = (L_ << 16);            // tensor_dim1 hi=0; tile_dim0 = 48 (bits 127:112)
  g1.s4 = ROWS_PER_BLK;          // tile_dim1 = 64; tile_dim2 = 0 (2D)
  g1.s5 = T_ * L_;               // tensor_dim0_stride = 24576 elements (row pitch)
  g1.s6 = 0;                     // stride hi / dim1_stride (unused, 2D)
  g1.s7 = 0;

  v4si g2 = {0, 0, 0, 0};        // unused higher dims
  v4si g3 = {0, 0, 0, 0};
  v8si g4 = {0, 0, 0, 0, 0, 0, 0, 0};  // 6-arg toolchain: extra group, zero

  __builtin_amdgcn_tensor_load_to_lds(g0, g1, g2, g3, g4, 0);
}

__global__ __launch_bounds__(WAVES_PER_BLK * 32)
void crf_fwd_wmma(const float* __restrict__ h,
                  const float* __restrict__ mask,
                  const float* __restrict__ trans,
                  float* __restrict__ out) {
  __shared__ float sc[ROWS_PER_BLK][SC_STRIDE];     // running scores
  __shared__ float pexp[ROWS_PER_BLK][SC_STRIDE];   // exp(score - rowmax)
  __shared__ float rmax_s[ROWS_PER_BLK];
  __shared__ float mrow_s[ROWS_PER_BLK];
  __shared__ float hbuf[2][HBUF_DW];                // TDM double buffer for h

  const int tid   = threadIdx.x;
  const int wave  = tid >> 5;
  const int lane  = tid & 31;
  const int m16   = lane & 15;
  const int hi    = lane >> 4;
  const int wrow0 = wave * ROWS_PER_WAVE;
  const int brow0 = blockIdx.x * ROWS_PER_BLK;      // block's first batch row
  const int grow0 = brow0 + wrow0;                  // this wave's first batch row

  // ---- kick off TDM for t = 0 into buffer 0 ------------------------------
  if (wave == 0) tdm_load_h_tile(h, brow0, 0, &hbuf[0][0]);

  // ---- Preload constant B-fragments: W[j][i] = exp(trans[i][j]) ----------
  v2f wB[12][3];
#pragma unroll
  for (int kt = 0; kt < 12; ++kt) {
#pragma unroll
    for (int nt = 0; nt < 3; ++nt) {
      const int n = nt * 16 + m16;          // target label i
      const int k = kt * 4 + 2 * hi;        // source label j
      wB[kt][nt].x = __expf(trans[n * L_ + k]);
      wB[kt][nt].y = __expf(trans[n * L_ + k + 1]);
    }
  }

  // ---- init: score0 = NEG everywhere except BOS = 0 ----------------------
#pragma unroll
  for (int jj = 0; jj < 24; ++jj) {
    const int j = hi * 24 + jj;
    sc[wrow0 + m16][j] = (j == BOS_) ? 0.0f : NEG_;
  }
  __builtin_amdgcn_wave_barrier();

  for (int t = 0; t < T_; ++t) {
    const int buf = t & 1;

    // issue TDM for next step into the other buffer (runs under this step)
    if (wave == 0 && (t + 1) < T_)
      tdm_load_h_tile(h, brow0, t + 1, &hbuf[buf ^ 1][0]);

    // 1) per-row max; lanes with hi==0 also stage this row's mask value
    const int myrow = wrow0 + m16;
    float mx = -3.0e38f;
#pragma unroll
    for (int jj = 0; jj < 24; ++jj)
      mx = fmaxf(mx, sc[myrow][hi * 24 + jj]);
    mx = fmaxf(mx, __shfl_xor(mx, 16, 32));
    if (hi == 0) {
      rmax_s[myrow] = mx;
      mrow_s[myrow] = mask[(size_t)(grow0 + m16) * T_ + t];
    }
    __builtin_amdgcn_wave_barrier();

    // 2) P = exp(score - rowmax)   (values in (0, 1])
#pragma unroll
    for (int jj = 0; jj < 24; ++jj) {
      const int j = hi * 24 + jj;
      pexp[myrow][j] = __expf(sc[myrow][j] - mx);
    }
    __builtin_amdgcn_wave_barrier();

    // 3) S = P @ W : (16x48)x(48x48), K chunks of 4, 3 N-tiles
    v8f acc0 = {0.f,0.f,0.f,0.f,0.f,0.f,0.f,0.f};
    v8f acc1 = acc0, acc2 = acc0;
#pragma unroll
    for (int kt = 0; kt < 12; ++kt) {
      v2f a;                                 // A-frag 16x4: M = m16, K = 2*hi+{0,1}
      const int k = kt * 4 + 2 * hi;
      a.x = pexp[wrow0 + m16][k];
      a.y = pexp[wrow0 + m16][k + 1];
      acc0 = __builtin_amdgcn_wmma_f32_16x16x4_f32(false, a, false, wB[kt][0],
                                                   (short)0, acc0, false, false);
      acc1 = __builtin_amdgcn_wmma_f32_16x16x4_f32(false, a, false, wB[kt][1],
                                                   (short)0, acc1, false, false);
      acc2 = __builtin_amdgcn_wmma_f32_16x16x4_f32(false, a, false, wB[kt][2],
                                                   (short)0, acc2, false, false);
    }

    // make this step's h tile visible: wave0 drains TENSORcnt down to the
    // in-flight t+1 load, then the block barrier publishes LDS to all waves
    if (wave == 0) {
      if ((t + 1) < T_) __builtin_amdgcn_s_wait_tensorcnt(1);
      else              __builtin_amdgcn_s_wait_tensorcnt(0);
    }
    __syncthreads();

    // 4) score_new = h_t + log(S) + rowmax; masked select; write back
    //    C/D layout: vgpr i, lane l -> row = i + 8*hi, col = nt*16 + m16
    //    staged h tile: addr = row*51 + nt*17 + m16 (odd stride, conflict-free)
#pragma unroll
    for (int i = 0; i < 8; ++i) {
      const int row  = i + 8 * hi;
      const int lrow = wrow0 + row;
      const float rmx  = rmax_s[lrow];
      const float mval = mrow_s[lrow];
      const float sv0 = acc0[i], sv1 = acc1[i], sv2 = acc2[i];
      {
        const float snew = hbuf[buf][lrow * HROW_STRIDE + 0 * 17 + m16]
                           + __logf(sv0) + rmx;
        sc[lrow][0 * 16 + m16] = (mval > 0.5f) ? snew : sc[lrow][0 * 16 + m16];
      }
      {
        const float snew = hbuf[buf][lrow * HROW_STRIDE + 1 * 17 + m16]
                           + __logf(sv1) + rmx;
        sc[lrow][1 * 16 + m16] = (mval > 0.5f) ? snew : sc[lrow][1 * 16 + m16];
      }
      {
        const float snew = hbuf[buf][lrow * HROW_STRIDE + 2 * 17 + m16]
                           + __logf(sv2) + rmx;
        sc[lrow][2 * 16 + m16] = (mval > 0.5f) ? snew : sc[lrow][2 * 16 + m16];
      }
    }
    // fence the just-read buffer before the t+2 TDM overwrites it
    __syncthreads();
  }

  // ---- final per-row logsumexp over the 48 labels ------------------------
  {
    const int row = wrow0 + m16;
    float mx = -3.0e38f;
#pragma unroll
    for (int jj = 0; jj < 24; ++jj)
      mx = fmaxf(mx, sc[row][hi * 24 + jj]);
    mx = fmaxf(mx, __shfl_xor(mx, 16, 32));
    float s = 0.0f;
#pragma unroll
    for (int jj = 0; jj < 24; ++jj)
      s += __expf(sc[row][hi * 24 + jj] - mx);
    s += __shfl_xor(s, 16, 32);
    if (hi == 0) out[grow0 + m16] = mx + __logf(s);
  }
}

extern "C" void kernel_launch(void* const* d_in, const int* in_sizes, int n_in,
                              void* d_out, int out_size, void* d_ws, size_t ws_size,
                              hipStream_t stream) {
  (void)in_sizes; (void)n_in; (void)d_ws; (void)ws_size; (void)out_size;
  const float* h     = (const float*)d_in[0];
  const float* mask  = (const float*)d_in[1];
  const float* trans = (const float*)d_in[2];
  float* out = (float*)d_out;

  dim3 grid(B_ / ROWS_PER_BLK);          // 16 blocks of 64 batch rows
  dim3 block(WAVES_PER_BLK * 32);        // 4 wave32 waves; 1 WMMA row-tile each
  crf_fwd_wmma<<<grid, block, 0, stream>>>(h, mask, trans, out);
}